// PointCNN_20177756356975
// MI455X (gfx1250) — compile-verified
//
#include <hip/hip_runtime.h>
#include <math.h>

// PointCNN forward for MI455X (gfx1250, wave32).
// All GEMM work goes through one wave-level WMMA kernel: each wave32 owns a
// 64x16 C tile (4 x v8f accumulators) and steps K by 4 with
// v_wmma_f32_16x16x4_f32.  All shapes in this net satisfy M%64==0 and K%4==0
// (d1's K=6 is zero-padded to 8), and N overflow is handled with branchless
// clamp+select.  The kernel is templated on (wTrans, act) and software-
// pipelined: next k-step fragments are loaded while the current 4 WMMAs
// execute, so the per-wave load->wmma chain is broken.  Whole working set
// < L2 (192MB): latency-bound.  KNN stages candidate tiles into LDS with
// global_load_async_to_lds_b32 + s_wait_asynccnt (CDNA5 async path).

typedef float v2f __attribute__((ext_vector_type(2)));
typedef float v8f __attribute__((ext_vector_type(8)));

__device__ __forceinline__ float elu1(float x) {
  return x > 0.f ? x : (__expf(x) - 1.f);
}

// ---------------------------------------------------------------------------
// WMMA f32 GEMM: C[M x N] = act(A[M x K] @ W + bias)
// Preconditions (all calls below satisfy them): M % 64 == 0, K % 4 == 0,
// lda/ldw even (8-byte aligned v2f fragment loads).
// A lane layout (16x4): lanes 0-15 -> M=lane, k=0..1; lanes 16-31 -> k=2..3.
// B lane layout (4x16): lanes 0-15 -> N=lane, k=0..1; lanes 16-31 -> k=2..3.
// C/D: reg r, lane l -> M = r + 8*(l>>4), N = l&15.
// WT=1 reads W as (N x K) row-major (xc_w 'odk' einsum).
// ---------------------------------------------------------------------------
template <int WT, int ACT>
__global__ void gemm_wmma_kernel(const float* __restrict__ A, int lda,
                                 const float* __restrict__ W, int ldw,
                                 const float* __restrict__ bias,
                                 float* __restrict__ C, int ldc,
                                 long M, int K, int N) {
  const int wave = threadIdx.x >> 5;
  const int lane = threadIdx.x & 31;
  const long tilesM = M >> 6;  // 64-row tiles, M%64==0
  const int tilesN = (N + 15) >> 4;
  long tile = (long)blockIdx.x * (blockDim.x >> 5) + wave;
  if (tile >= tilesM * (long)tilesN) return;  // wave-uniform exit
  const long tm = tile / tilesN;
  const int tn = (int)(tile % tilesN);
  const int lrow = lane & 15;
  const int kh = (lane >> 4) << 1;
  const int colRaw = tn * 16 + lrow;
  const bool colOK = colRaw < N;
  const int col = colOK ? colRaw : (N - 1);  // clamped address, value zeroed

  const float* Ar = A + (tm * 64 + lrow) * (long)lda + kh;
  const float* Bp = WT ? (W + (long)col * ldw + kh)
                       : (W + (long)kh * ldw + col);
  const long bStep = WT ? 4 : 4L * ldw;

  auto loadB = [&](const float* p) -> v2f {
    v2f r;
    if (WT) {
      r = *(const v2f*)p;
    } else {
      r.x = p[0];
      r.y = p[ldw];
    }
    if (!colOK) { r.x = 0.f; r.y = 0.f; }
    return r;
  };

  v8f acc[4] = {v8f{}, v8f{}, v8f{}, v8f{}};
  // software pipeline: current fragments live while next loads are in flight
  v2f ca[4], cb;
  cb = loadB(Bp);
#pragma unroll
  for (int mt = 0; mt < 4; ++mt)
    ca[mt] = *(const v2f*)(Ar + (long)mt * 16 * lda);

  for (int k0 = 0; k0 + 4 < K; k0 += 4) {
    Bp += bStep;
    Ar += 4;
    v2f nb = loadB(Bp);
    v2f na[4];
#pragma unroll
    for (int mt = 0; mt < 4; ++mt)
      na[mt] = *(const v2f*)(Ar + (long)mt * 16 * lda);
#pragma unroll
    for (int mt = 0; mt < 4; ++mt)
      acc[mt] = __builtin_amdgcn_wmma_f32_16x16x4_f32(false, ca[mt], false, cb,
                                                      (short)0, acc[mt],
                                                      false, false);
    cb = nb;
#pragma unroll
    for (int mt = 0; mt < 4; ++mt) ca[mt] = na[mt];
  }
#pragma unroll
  for (int mt = 0; mt < 4; ++mt)
    acc[mt] = __builtin_amdgcn_wmma_f32_16x16x4_f32(false, ca[mt], false, cb,
                                                    (short)0, acc[mt], false,
                                                    false);

  if (colOK) {
    const float bv = bias ? bias[colRaw] : 0.f;
    const int rbase = (lane >> 4) << 3;
#pragma unroll
    for (int mt = 0; mt < 4; ++mt) {
      const long cm0 = tm * 64 + mt * 16 + rbase;
#pragma unroll
      for (int r = 0; r < 8; ++r) {
        float v = acc[mt][r] + bv;
        if (ACT) v = elu1(v);
        C[(cm0 + r) * ldc + colRaw] = v;
      }
    }
  }
}

// ---------------------------------------------------------------------------
// x (B,6,N) -> pts (B,N,6)
// ---------------------------------------------------------------------------
__global__ void transpose_x_kernel(const float* __restrict__ x,
                                   float* __restrict__ pts, int N) {
  long i = (long)blockIdx.x * blockDim.x + threadIdx.x;  // over 8*6*N
  if (i >= 8L * 6 * N) return;
  int n = (int)(i % N);
  int d = (int)((i / N) % 6);
  long b = i / (6L * N);
  pts[(b * N + n) * 6 + d] = x[i];
}

// Deterministic sub-sampling (stands in for np.random permutation; shape-exact).
__global__ void sample_kernel(const float* __restrict__ pts,
                              float* __restrict__ rep, int N, int P) {
  long i = (long)blockIdx.x * blockDim.x + threadIdx.x;  // over 8*P
  if (i >= 8L * P) return;
  int p = (int)(i % P);
  long b = i / P;
  unsigned j = ((unsigned)p * 2654435761u + 12345u) % (unsigned)N;
#pragma unroll
  for (int d = 0; d < 6; ++d) rep[i * 6 + d] = pts[((long)b * N + j) * 6 + d];
}

// Zero-pad a (rows x cols) row-major matrix to (rowsPad x cols).
__global__ void pad_rows_kernel(const float* __restrict__ src,
                                float* __restrict__ dst, int rows, int rowsPad,
                                int cols) {
  int t = blockIdx.x * blockDim.x + threadIdx.x;
  if (t >= rowsPad * cols) return;
  int r = t / cols;
  dst[t] = (r < rows) ? src[t] : 0.f;
}

// ---------------------------------------------------------------------------
// KNN: per rep point keep the (K*D+1) nearest (register insertion sort, fully
// unrolled -> VGPRs).  Candidate tiles are staged into LDS with the CDNA5
// async Global->LDS DMA path (ASYNCcnt), shared by the whole block.
// ---------------------------------------------------------------------------
template <int KD1, int KK, int DD>
__global__ void knn_kernel(const float* __restrict__ rep,
                           const float* __restrict__ pts,
                           int* __restrict__ nb, int Pn, int N) {
  __shared__ float tile[128 * 6];
  const int b = blockIdx.y;
  const int p = blockIdx.x * blockDim.x + threadIdx.x;
  const bool active = p < Pn;
  float rx[6];
#pragma unroll
  for (int d = 0; d < 6; ++d)
    rx[d] = active ? rep[((long)b * Pn + p) * 6 + d] : 0.f;
  float bd[KD1];
  int bi[KD1];
#pragma unroll
  for (int s = 0; s < KD1; ++s) { bd[s] = 3.0e38f; bi[s] = 0; }

  for (int n0 = 0; n0 < N; n0 += 128) {
    const int cnt = min(128, N - n0);
    __syncthreads();
    // async DMA: each lane pushes dwords global -> LDS, no VGPR round-trip
    for (int t = threadIdx.x; t < cnt * 6; t += blockDim.x) {
      unsigned lds = (unsigned)(uintptr_t)&tile[t];
      const float* g = &pts[((long)b * N + n0) * 6 + t];
      asm volatile("global_load_async_to_lds_b32 %0, %1, off"
                   :
                   : "v"(lds), "v"(g)
                   : "memory");
    }
    asm volatile("s_wait_asynccnt 0" ::: "memory");
    __syncthreads();
    if (active) {
      for (int j = 0; j < cnt; ++j) {
        float d2 = 0.f;
#pragma unroll
        for (int d = 0; d < 6; ++d) {
          float df = rx[d] - tile[j * 6 + d];
          d2 = fmaf(df, df, d2);
        }
        if (d2 < bd[KD1 - 1]) {
          bd[KD1 - 1] = d2;
          bi[KD1 - 1] = n0 + j;
#pragma unroll
          for (int s = KD1 - 1; s > 0; --s) {
            if (bd[s] < bd[s - 1]) {
              float td = bd[s]; bd[s] = bd[s - 1]; bd[s - 1] = td;
              int ti = bi[s]; bi[s] = bi[s - 1]; bi[s - 1] = ti;
            } else break;
          }
        }
      }
    }
  }
  if (active) {
#pragma unroll
    for (int k = 0; k < KK; ++k)
      nb[((long)b * Pn + p) * KK + k] = bi[1 + k * DD];
  }
}

// ---------------------------------------------------------------------------
// Gather: local = pts[nb] - rep.  localK is stride-8 zero-padded (so the d1
// GEMM runs K=8 guard-free); localT is (6,K)-major for the xc einsum-as-GEMM.
// Also scatters dense-transformed features into fcat's tail columns.
// ---------------------------------------------------------------------------
__global__ void gather_local_kernel(const float* __restrict__ rep,
                                    const float* __restrict__ pts,
                                    const int* __restrict__ nb,
                                    const float* __restrict__ fprev,
                                    float* __restrict__ localK,
                                    float* __restrict__ localT,
                                    float* __restrict__ fcat, int Pn, int N,
                                    int K, int Cmid, int Cprev, int Ccat,
                                    long total) {
  long i = (long)blockIdx.x * blockDim.x + threadIdx.x;  // over B*Pn*K
  if (i >= total) return;
  const int k = (int)(i % K);
  const long bp = i / K;  // b*Pn + p
  const int n = nb[i];
  const long b = bp / Pn;
#pragma unroll
  for (int d = 0; d < 6; ++d) {
    float v = pts[((long)b * N + n) * 6 + d] - rep[bp * 6 + d];
    localK[i * 8 + d] = v;
    localT[(bp * 6 + d) * K + k] = v;
  }
  localK[i * 8 + 6] = 0.f;
  localK[i * 8 + 7] = 0.f;
  if (fprev) {
    for (int c = 0; c < Cprev; ++c)
      fcat[i * Ccat + Cmid + c] = fprev[((long)b * N + n) * Cprev + c];
  }
}

// ---------------------------------------------------------------------------
// Per point: fx = X(KxK) @ fcat(KxCcat), then depthwise
// dw[c*dm+m] = sum_k fx[k,c] * dw_w[c,m,k] + dw_b.  Staged through LDS.
// ---------------------------------------------------------------------------
__global__ void fxdw_kernel(const float* __restrict__ X,
                            const float* __restrict__ fcat,
                            const float* __restrict__ dw_w,
                            const float* __restrict__ dw_b,
                            float* __restrict__ dwout, int K, int Ccat,
                            int dm) {
  __shared__ float sX[144];   // K*K <= 144
  __shared__ float sF[1152];  // K*Ccat <= 12*96
  __shared__ float sfx[1152];
  const long p = blockIdx.x;
  const int K2 = K * K, KC = K * Ccat, CD = Ccat * dm;
  for (int t = threadIdx.x; t < K2; t += blockDim.x) sX[t] = X[p * K2 + t];
  for (int t = threadIdx.x; t < KC; t += blockDim.x) sF[t] = fcat[p * KC + t];
  __syncthreads();
  for (int t = threadIdx.x; t < KC; t += blockDim.x) {
    const int i = t / Ccat, c = t % Ccat;
    float s = 0.f;
    for (int j = 0; j < K; ++j) s = fmaf(sX[i * K + j], sF[j * Ccat + c], s);
    sfx[t] = s;
  }
  __syncthreads();
  for (int t = threadIdx.x; t < CD; t += blockDim.x) {
    const int c = t / dm;
    float s = 0.f;
    for (int k = 0; k < K; ++k)
      s = fmaf(sfx[k * Ccat + c], dw_w[(long)t * K + k], s);
    dwout[p * CD + t] = s + dw_b[t];
  }
}

// ---------------------------------------------------------------------------
// BatchNorm: LDS per-channel partials, one global atomic per channel/block.
// stats[0..C)=sum, stats[C..2C)=sumsq.
// ---------------------------------------------------------------------------
__global__ void zero_kernel(float* __restrict__ p, int n) {
  for (int t = threadIdx.x; t < n; t += blockDim.x) p[t] = 0.f;
}

__global__ void bn_stats_kernel(const float* __restrict__ A,
                                float* __restrict__ stats, long M, int C) {
  __shared__ float s[256];  // 2*C <= 256
  for (int t = threadIdx.x; t < 2 * C; t += blockDim.x) s[t] = 0.f;
  __syncthreads();
  const long total = M * (long)C;
  const long base = (long)blockIdx.x * blockDim.x * 16;
  for (int r = 0; r < 16; ++r) {
    long i = base + (long)r * blockDim.x + threadIdx.x;
    if (i < total) {
      float v = A[i];
      int c = (int)(i % C);
      atomicAdd(&s[c], v);
      atomicAdd(&s[C + c], v * v);
    }
  }
  __syncthreads();
  for (int t = threadIdx.x; t < 2 * C; t += blockDim.x)
    atomicAdd(&stats[t], s[t]);
}

__global__ void bn_apply_kernel(const float* __restrict__ A,
                                const float* __restrict__ stats,
                                const float* __restrict__ g,
                                const float* __restrict__ bta,
                                float* __restrict__ out, long M, int C,
                                int act) {
  long i = (long)blockIdx.x * blockDim.x + threadIdx.x;
  if (i >= M * (long)C) return;
  const int c = (int)(i % C);
  const float inv = 1.f / (float)M;
  const float mean = stats[c] * inv;
  const float var = stats[C + c] * inv - mean * mean;
  float v = (A[i] - mean) * rsqrtf(var + 1e-5f) * g[c] + bta[c];
  out[i] = act ? elu1(v) : v;
}

// ---------------------------------------------------------------------------
// Head: out[b,o,p] = sum_c h[b,p,c] * w[o,c]   (h is (B,P,32))
// ---------------------------------------------------------------------------
__global__ void heads_kernel(const float* __restrict__ h,
                             const float* __restrict__ w,
                             float* __restrict__ out, int nc, int P,
                             long total) {
  long i = (long)blockIdx.x * blockDim.x + threadIdx.x;
  if (i >= total) return;
  const int p = (int)(i % P);
  const int o = (int)((i / P) % nc);
  const long b = i / ((long)P * nc);
  float s = 0.f;
#pragma unroll
  for (int c = 0; c < 32; ++c)
    s = fmaf(h[(b * P + p) * 32 + c], w[o * 32 + c], s);
  out[i] = s;
}

// ===========================================================================
// Host driver
// ===========================================================================
namespace {

struct LP {
  const float *bn_b, *bn_g, *d1_b, *d1_w, *d2_b, *d2_w, *dense_b, *dense_w,
      *dw_b, *dw_w, *pw_w, *xc_b, *xc_w, *xd1_b, *xd1_w, *xd2_b, *xd2_w;
};

struct Cfg {
  int Cin, Cout, K, D, P, Cmid, Cprev, dm, Ccat;
};

const Cfg kCfgs[7] = {
    {0, 32, 8, 1, -1, 16, 0, 1, 16},      {32, 64, 8, 2, 2048, 16, 32, 2, 48},
    {64, 96, 8, 4, 512, 24, 48, 2, 72},   {96, 128, 12, 4, 128, 32, 64, 2, 96},
    {128, 96, 12, 4, 256, 24, 48, 1, 72}, {96, 64, 8, 4, 512, 16, 32, 1, 48},
    {64, 32, 8, 2, 2048, 8, 16, 1, 24},
};

void gemm(const float* A, int lda, const float* W, int ldw, int wT,
          const float* bias, float* C, int ldc, long M, int K, int N, int act,
          hipStream_t stream) {
  const long tilesM = M / 64;  // M % 64 == 0 for every call in this net
  const long tilesN = (N + 15) / 16;
  const long tiles = tilesM * tilesN;
  const long blocks = (tiles + 7) / 8;  // 8 waves per block
  dim3 g((unsigned)blocks), blk(256);
  if (wT) {
    if (act)
      gemm_wmma_kernel<1, 1><<<g, blk, 0, stream>>>(A, lda, W, ldw, bias, C,
                                                    ldc, M, K, N);
    else
      gemm_wmma_kernel<1, 0><<<g, blk, 0, stream>>>(A, lda, W, ldw, bias, C,
                                                    ldc, M, K, N);
  } else {
    if (act)
      gemm_wmma_kernel<0, 1><<<g, blk, 0, stream>>>(A, lda, W, ldw, bias, C,
                                                    ldc, M, K, N);
    else
      gemm_wmma_kernel<0, 0><<<g, blk, 0, stream>>>(A, lda, W, ldw, bias, C,
                                                    ldc, M, K, N);
  }
}

}  // namespace

extern "C" void kernel_launch(void* const* d_in, const int* in_sizes, int n_in,
                              void* d_out, int out_size, void* d_ws,
                              size_t ws_size, hipStream_t stream) {
  (void)out_size;
  (void)ws_size;
  auto F = [&](int i) { return (const float*)d_in[i]; };

  // ----- map harness inputs (support both flattening conventions) -----
  LP L[7];
  const float *top_bn_g = nullptr, *top_bn_b = nullptr, *x = nullptr;
  const float* headsW[5] = {nullptr, nullptr, nullptr, nullptr, nullptr};
  const bool xFirst = (in_sizes && n_in > 0 && in_sizes[0] == 8 * 6 * 4096);
  if (xFirst) {
    // insertion order: x, l0..l6 (creation order), bn_g, bn_b, heads
    x = F(0);
    int idx = 1;
    for (int i = 0; i < 7; ++i) {
      LP& l = L[i];
      if (i > 0) { l.dense_w = F(idx++); l.dense_b = F(idx++); }
      else       { l.dense_w = nullptr;  l.dense_b = nullptr; }
      l.d1_w = F(idx++); l.d1_b = F(idx++);
      l.d2_w = F(idx++); l.d2_b = F(idx++);
      l.xc_w = F(idx++); l.xc_b = F(idx++);
      l.xd1_w = F(idx++); l.xd1_b = F(idx++);
      l.xd2_w = F(idx++); l.xd2_b = F(idx++);
      l.dw_w = F(idx++); l.dw_b = F(idx++);
      l.pw_w = F(idx++); l.bn_g = F(idx++); l.bn_b = F(idx++);
    }
    top_bn_g = F(idx++); top_bn_b = F(idx++);
    for (int h = 0; h < 5; ++h) headsW[h] = F(idx++);
  } else {
    // jax tree order: sorted dict keys -> params first, then x
    top_bn_b = F(0); top_bn_g = F(1);
    for (int h = 0; h < 5; ++h) headsW[h] = F(2 + h);
    int idx = 7;
    for (int i = 0; i < 7; ++i) {
      LP& l = L[i];
      l.bn_b = F(idx++); l.bn_g = F(idx++);
      l.d1_b = F(idx++); l.d1_w = F(idx++);
      l.d2_b = F(idx++); l.d2_w = F(idx++);
      if (i > 0) { l.dense_b = F(idx++); l.dense_w = F(idx++); }
      else       { l.dense_b = nullptr;  l.dense_w = nullptr; }
      l.dw_b = F(idx++); l.dw_w = F(idx++); l.pw_w = F(idx++);
      l.xc_b = F(idx++); l.xc_w = F(idx++);
      l.xd1_b = F(idx++); l.xd1_w = F(idx++);
      l.xd2_b = F(idx++); l.xd2_w = F(idx++);
    }
    x = F(idx);
  }

  // ----- workspace carve-up (floats) -----
  char* wsb = (char*)d_ws;
  size_t off = 0;
  auto allocf = [&](size_t nf) {
    void* r = wsb + off;
    off += ((nf * 4 + 255) & ~(size_t)255);
    return (float*)r;
  };
  float* pts0 = allocf(196608);    // 8*4096*6
  float* pts1 = allocf(196608);
  float* ftsB = allocf(1048576);   // max B*Pn*Cout
  float* dense = allocf(1048576);  // max B*N*Cprev (also reused as final h)
  float* lK = allocf(2097152);     // B*Pn*K*8 (stride-8 padded)
  float* lT = allocf(1572864);     // B*Pn*6*K
  float* hB = allocf(4194304);     // B*Pn*K*Cmid
  float* fcat = allocf(6291456);   // B*Pn*K*Ccat
  float* X1 = allocf(2097152);     // B*Pn*K^2
  float* X2 = allocf(2097152);
  float* dwB = allocf(1572864);    // B*Pn*Ccat*dm
  float* outB = allocf(1048576);   // B*Pn*Cout
  float* stats = allocf(256);
  float* d1pad = allocf(256);      // 8 x Cmid zero-padded d1_w
  int* nbB = (int*)allocf(393216); // B*Pn*K

  const int B = 8;

  auto bn = [&](const float* A, const float* g_, const float* b_, float* out,
                long M, int C, int act) {
    zero_kernel<<<1, 256, 0, stream>>>(stats, 2 * C);
    const long total = M * (long)C;
    const long sblocks = (total + 256 * 16 - 1) / (256 * 16);
    bn_stats_kernel<<<(unsigned)sblocks, 256, 0, stream>>>(A, stats, M, C);
    bn_apply_kernel<<<(unsigned)((total + 255) / 256), 256, 0, stream>>>(
        A, stats, g_, b_, out, M, C, act);
  };

  // pts = transpose(x)
  {
    const long tot = 8L * 6 * 4096;
    transpose_x_kernel<<<(unsigned)((tot + 255) / 256), 256, 0, stream>>>(
        x, pts0, 4096);
  }

  int N = 4096;
  float* ptsP = pts0;
  const float* fts = nullptr;

  for (int li = 0; li < 7; ++li) {
    const Cfg& c = kCfgs[li];
    const LP& l = L[li];
    float* rep;
    int Pn;
    if (c.P > 0 && c.P < N) {
      rep = (ptsP == pts0) ? pts1 : pts0;
      Pn = c.P;
      const long tot = 8L * Pn;
      sample_kernel<<<(unsigned)((tot + 255) / 256), 256, 0, stream>>>(
          ptsP, rep, N, Pn);
    } else {
      rep = ptsP;
      Pn = N;
    }

    if (c.Cin > 0) {  // fts = elu(fts @ dense_w + dense_b) over all N points
      gemm(fts, c.Cin, l.dense_w, c.Cprev, 0, l.dense_b, dense, c.Cprev,
           (long)B * N, c.Cin, c.Cprev, 1, stream);
    }

    {  // KNN
      dim3 g((Pn + 127) / 128, B), blk(128);
      if (c.K == 8 && c.D == 1)
        knn_kernel<9, 8, 1><<<g, blk, 0, stream>>>(rep, ptsP, nbB, Pn, N);
      else if (c.K == 8 && c.D == 2)
        knn_kernel<17, 8, 2><<<g, blk, 0, stream>>>(rep, ptsP, nbB, Pn, N);
      else if (c.K == 8 && c.D == 4)
        knn_kernel<33, 8, 4><<<g, blk, 0, stream>>>(rep, ptsP, nbB, Pn, N);
      else
        knn_kernel<49, 12, 4><<<g, blk, 0, stream>>>(rep, ptsP, nbB, Pn, N);
    }

    const long totG = (long)B * Pn * c.K;
    gather_local_kernel<<<(unsigned)((totG + 255) / 256), 256, 0, stream>>>(
        rep, ptsP, nbB, c.Cin > 0 ? dense : nullptr, lK, lT, fcat, Pn, N, c.K,
        c.Cmid, c.Cprev, c.Ccat, totG);

    // zero-pad d1_w (6 x Cmid) to 8 rows so the d1 GEMM runs K=8 guard-free
    pad_rows_kernel<<<1, 256, 0, stream>>>(l.d1_w, d1pad, 6, 8, c.Cmid);

    // h = elu(local @ d1_w + b);  h = elu(h @ d2_w + b) -> fcat[:, :Cmid]
    gemm(lK, 8, d1pad, c.Cmid, 0, l.d1_b, hB, c.Cmid, totG, 8, c.Cmid, 1,
         stream);
    gemm(hB, c.Cmid, l.d2_w, c.Cmid, 0, l.d2_b, fcat, c.Ccat, totG, c.Cmid,
         c.Cmid, 1, stream);

    // X-transform chain (xc_w consumed transposed: W'[dk, o] = xc_w[o, d, k])
    const int K2 = c.K * c.K, K6 = 6 * c.K;
    gemm(lT, K6, l.xc_w, K6, 1, l.xc_b, X1, K2, (long)B * Pn, K6, K2, 1,
         stream);
    gemm(X1, K2, l.xd1_w, K2, 0, l.xd1_b, X2, K2, (long)B * Pn, K2, K2, 1,
         stream);
    gemm(X2, K2, l.xd2_w, K2, 0, l.xd2_b, X1, K2, (long)B * Pn, K2, K2, 0,
         stream);

    // fx + depthwise, then pointwise, then BN + ELU -> fts
    fxdw_kernel<<<(unsigned)(B * Pn), 128, 0, stream>>>(X1, fcat, l.dw_w,
                                                        l.dw_b, dwB, c.K,
                                                        c.Ccat, c.dm);
    gemm(dwB, c.Ccat * c.dm, l.pw_w, c.Cout, 0, nullptr, outB, c.Cout,
         (long)B * Pn, c.Ccat * c.dm, c.Cout, 0, stream);
    bn(outB, l.bn_g, l.bn_b, ftsB, (long)B * Pn, c.Cout, 1);

    fts = ftsB;
    ptsP = rep;
    N = Pn;
  }

  // final BN (no ELU) -> reuse `dense` as h, then heads
  bn(ftsB, top_bn_g, top_bn_b, dense, (long)B * N, 32, 0);
  static const int ncs[5] = {2, 4, 8, 16, 24};
  long oOff = 0;
  for (int h = 0; h < 5; ++h) {
    const long tot = (long)B * ncs[h] * N;
    heads_kernel<<<(unsigned)((tot + 255) / 256), 256, 0, stream>>>(
        dense, headsW[h], (float*)d_out + oOff, ncs[h], N, tot);
    oOff += tot;
  }
}